// ModelNew_1580547973039
// MI455X (gfx1250) — compile-verified
//
#include <hip/hip_runtime.h>
#include <hip/hip_bf16.h>

typedef __attribute__((ext_vector_type(2))) float v2f;
typedef __attribute__((ext_vector_type(8))) float v8f;

#define N_BATCH 256
#define C_IN    128
#define C_OUT   512
#define HW      4096          // 64*64
#define DENOM   16900.0f      // (63*2+4)^2 = 130*130
#define LN_EPS  1e-5f

__device__ __forceinline__ float wave_reduce_sum(float v) {
    v += __shfl_xor(v, 16, 32);
    v += __shfl_xor(v, 8, 32);
    v += __shfl_xor(v, 4, 32);
    v += __shfl_xor(v, 2, 32);
    v += __shfl_xor(v, 1, 32);
    return v;
}

// ---------------------------------------------------------------------------
// Kernel 1: xs[n*C_IN + c] = sum over 4096 spatial elements of plane (n,c).
// One 256-thread block per plane; fully coalesced float4 streaming loads.
// This kernel carries 536 MB of HBM traffic and bounds total runtime.
// ---------------------------------------------------------------------------
__global__ void reduce_x_kernel(const float* __restrict__ x, float* __restrict__ xs) {
    const int plane = blockIdx.x;                     // n*C_IN + c
    const float4* __restrict__ p4 =
        (const float4*)(x + (size_t)plane * HW);      // 1024 float4 per plane
    const int tid = threadIdx.x;

    float s = 0.0f;
#pragma unroll
    for (int i = 0; i < 4; ++i) {
        float4 v = p4[tid + 256 * i];                 // coalesced: lanes contiguous
        s += v.x + v.y + v.z + v.w;
    }
    s = wave_reduce_sum(s);

    __shared__ float parts[8];
    const int wave = tid >> 5;
    const int lane = tid & 31;
    if (lane == 0) parts[wave] = s;
    __syncthreads();
    if (tid == 0) {
        float t = 0.0f;
#pragma unroll
        for (int i = 0; i < 8; ++i) t += parts[i];
        xs[plane] = t;
    }
}

// ---------------------------------------------------------------------------
// Kernel 2: wsum[ci*C_OUT + co] = sum of 16 contiguous kernel taps.
// weight layout [C_IN, C_OUT, 4, 4] -> taps for (ci,co) are contiguous.
// ---------------------------------------------------------------------------
__global__ void reduce_w_kernel(const float* __restrict__ w, float* __restrict__ wsum) {
    const int idx = blockIdx.x * 256 + threadIdx.x;   // ci*C_OUT + co, 0..65535
    const float4* __restrict__ p = (const float4*)(w + (size_t)idx * 16);
    float4 a = p[0], b = p[1], c = p[2], d = p[3];
    wsum[idx] = (a.x + a.y + a.z + a.w) + (b.x + b.y + b.z + b.w)
              + (c.x + c.y + c.z + c.w) + (d.x + d.y + d.z + d.w);
}

// ---------------------------------------------------------------------------
// Kernel 3: y0 = xs @ wsum / DENOM + bias, then LN -> clip(-1,1) -> LN per row.
// Block = 256 threads (8 waves) owns 16 rows x 512 cols so both LayerNorms
// (channel-dim, i.e. per output row) fuse in-block.
// GEMM via V_WMMA_F32_16X16X4_F32 (exact f32): 8 waves x 4 N-tiles x 32 K-steps.
// ---------------------------------------------------------------------------
__global__ void gemm_ln_kernel(const float* __restrict__ xs,
                               const float* __restrict__ wsum,
                               const float* __restrict__ bias,
                               const float* __restrict__ g1, const float* __restrict__ b1,
                               const float* __restrict__ g2, const float* __restrict__ b2,
                               float* __restrict__ out) {
    __shared__ float sA[16 * C_IN];    // xs tile: 16 rows x 128 (8 KB)
    __shared__ float sY[16 * C_OUT];   // y0 rows: 16 x 512 (32 KB)

    const int tid  = threadIdx.x;      // 0..255
    const int wave = tid >> 5;         // 0..7
    const int lane = tid & 31;
    const int row0 = blockIdx.x * 16;

    // Stage the 16x128 xs tile into LDS (2048 floats, float4-coalesced).
    {
        const float4* src = (const float4*)(xs + (size_t)row0 * C_IN);
        float4* dst = (float4*)sA;
        for (int i = tid; i < 16 * C_IN / 4; i += 256) dst[i] = src[i];
    }
    __syncthreads();

    // WMMA f32 16x16x4 fragment addressing (per ISA VGPR layout tables):
    //  A (16x4): lanes 0-15 hold K=0,1 ; lanes 16-31 hold K=2,3 ; M = lane&15
    //  B (4x16): lanes 0-15 hold K=0,1 ; lanes 16-31 hold K=2,3 ; N = lane&15
    //  D (16x16): col = lane&15 ; row = (lane>>4)*8 + vgpr
    const int m16   = lane & 15;
    const int khalf = (lane >> 4) * 2;  // 0 or 2

    v8f acc[4] = {};                    // 4 N-tiles per wave, f32 accumulators
    for (int kk = 0; kk < 32; ++kk) {
        const int kb = kk * 4 + khalf;
        v2f a;
        a.x = sA[m16 * C_IN + kb + 0];
        a.y = sA[m16 * C_IN + kb + 1];
#pragma unroll
        for (int t = 0; t < 4; ++t) {
            const int n0 = (wave * 4 + t) * 16;
            v2f bf;
            bf.x = wsum[(size_t)(kb + 0) * C_OUT + n0 + m16];
            bf.y = wsum[(size_t)(kb + 1) * C_OUT + n0 + m16];
            acc[t] = __builtin_amdgcn_wmma_f32_16x16x4_f32(
                false, a, false, bf, (short)0, acc[t], false, false);
        }
    }

    // Epilogue of GEMM: scale + bias into LDS row buffer.
    const float inv_denom = 1.0f / DENOM;
#pragma unroll
    for (int t = 0; t < 4; ++t) {
        const int col   = (wave * 4 + t) * 16 + m16;
        const int rbase = (lane >> 4) * 8;
        const float bs  = bias[col];
#pragma unroll
        for (int r = 0; r < 8; ++r) {
            sY[(rbase + r) * C_OUT + col] = acc[t][r] * inv_denom + bs;
        }
    }
    __syncthreads();

    // LN -> hardtanh -> LN, one wave handles 2 rows, 16 channels per lane.
    for (int rr = 0; rr < 2; ++rr) {
        const int row = wave * 2 + rr;
        float v[16];
        float s = 0.0f;
#pragma unroll
        for (int i = 0; i < 16; ++i) {
            v[i] = sY[row * C_OUT + lane * 16 + i];
            s += v[i];
        }
        s = wave_reduce_sum(s);
        const float mu = s * (1.0f / C_OUT);
        float sq = 0.0f;
#pragma unroll
        for (int i = 0; i < 16; ++i) { float d = v[i] - mu; sq += d * d; }
        sq = wave_reduce_sum(sq);
        const float rstd = rsqrtf(sq * (1.0f / C_OUT) + LN_EPS);

        float s2 = 0.0f;
#pragma unroll
        for (int i = 0; i < 16; ++i) {
            const int c = lane * 16 + i;
            float y1 = (v[i] - mu) * rstd * g1[c] + b1[c];
            y1 = fminf(1.0f, fmaxf(-1.0f, y1));
            v[i] = y1;
            s2 += y1;
        }
        s2 = wave_reduce_sum(s2);
        const float mu2 = s2 * (1.0f / C_OUT);
        float sq2 = 0.0f;
#pragma unroll
        for (int i = 0; i < 16; ++i) { float d = v[i] - mu2; sq2 += d * d; }
        sq2 = wave_reduce_sum(sq2);
        const float rstd2 = rsqrtf(sq2 * (1.0f / C_OUT) + LN_EPS);

        float* orow = out + (size_t)(row0 + row) * C_OUT;
#pragma unroll
        for (int i = 0; i < 16; ++i) {
            const int c = lane * 16 + i;
            orow[c] = (v[i] - mu2) * rstd2 * g2[c] + b2[c];
        }
    }
}

extern "C" void kernel_launch(void* const* d_in, const int* in_sizes, int n_in,
                              void* d_out, int out_size, void* d_ws, size_t ws_size,
                              hipStream_t stream) {
    (void)in_sizes; (void)n_in; (void)out_size; (void)ws_size;
    const float* x      = (const float*)d_in[0];
    const float* weight = (const float*)d_in[1];
    const float* bias   = (const float*)d_in[2];
    const float* g1     = (const float*)d_in[3];
    const float* b1     = (const float*)d_in[4];
    const float* g2     = (const float*)d_in[5];
    const float* b2     = (const float*)d_in[6];

    float* xs   = (float*)d_ws;                 // [256, 128]  = 128 KB
    float* wsum = xs + N_BATCH * C_IN;          // [128, 512]  = 256 KB

    reduce_x_kernel<<<N_BATCH * C_IN, 256, 0, stream>>>(x, xs);
    reduce_w_kernel<<<(C_IN * C_OUT) / 256, 256, 0, stream>>>(weight, wsum);
    gemm_ln_kernel<<<N_BATCH / 16, 256, 0, stream>>>(xs, wsum, bias, g1, b1, g2, b2,
                                                     (float*)d_out);
}